// EUNNLayer_39221641347703
// MI455X (gfx1250) — compile-verified
//
#include <hip/hip_runtime.h>
#include <math.h>

#define H      1024
#define LAYERS 32
#define ROWS   8
#define PAIRS  (ROWS / 2)
#define TPB    256

typedef float f2 __attribute__((ext_vector_type(2)));

#if defined(__AMDGCN__) && __has_builtin(__builtin_amdgcn_global_load_async_to_lds_b128)
#define USE_ASYNC 1
#define AS_GLOBAL __attribute__((address_space(1)))
#define AS_LDS    __attribute__((address_space(3)))
typedef int v4i __attribute__((vector_size(16)));
#else
#define USE_ASYNC 0
#endif

// ---- packed fp32 helpers: d = a * splat(b.lo/hi) (+/- ) + c ------------------
// op_sel picks the lo/hi half of the 64-bit coefficient pair for BOTH result
// lanes (free splat); neg_lo/neg_hi fold the complex-multiply sign pattern.
#if defined(__HIP_DEVICE_COMPILE__) && defined(__AMDGCN__)
#define PKFMA(name, mods)                                                  \
  __device__ __forceinline__ f2 name(f2 a, f2 b, f2 c) {                   \
    f2 d;                                                                  \
    asm("v_pk_fma_f32 %0, %1, %2, %3 " mods                                \
        : "=v"(d) : "v"(a), "v"(b), "v"(c));                               \
    return d;                                                              \
  }
#define PKMUL(name, mods)                                                  \
  __device__ __forceinline__ f2 name(f2 a, f2 b) {                         \
    f2 d;                                                                  \
    asm("v_pk_mul_f32 %0, %1, %2 " mods                                    \
        : "=v"(d) : "v"(a), "v"(b));                                       \
    return d;                                                              \
  }
PKFMA(pk_fma_blo,  "op_sel:[0,0,0] op_sel_hi:[1,0,1]")                      // a*b.lo + c
PKFMA(pk_fma_bhi,  "op_sel:[0,1,0] op_sel_hi:[1,1,1]")                      // a*b.hi + c
PKFMA(pk_fma_nbhi, "op_sel:[0,1,0] op_sel_hi:[1,1,1] neg_lo:[0,1,0] neg_hi:[0,1,0]") // -a*b.hi + c
PKMUL(pk_mul_blo,  "op_sel:[0,0] op_sel_hi:[1,0]")                          // a*b.lo
PKMUL(pk_mul_nbhi, "op_sel:[0,1] op_sel_hi:[1,1] neg_lo:[0,1] neg_hi:[0,1]") // -a*b.hi
#else
__device__ __forceinline__ f2 pk_fma_blo(f2 a, f2 b, f2 c)  { f2 s = {b.x,  b.x};  return __builtin_elementwise_fma(a, s, c); }
__device__ __forceinline__ f2 pk_fma_bhi(f2 a, f2 b, f2 c)  { f2 s = {b.y,  b.y};  return __builtin_elementwise_fma(a, s, c); }
__device__ __forceinline__ f2 pk_fma_nbhi(f2 a, f2 b, f2 c) { f2 s = {-b.y, -b.y}; return __builtin_elementwise_fma(a, s, c); }
__device__ __forceinline__ f2 pk_mul_blo(f2 a, f2 b)        { f2 s = {b.x,  b.x};  return a * s; }
__device__ __forceinline__ f2 pk_mul_nbhi(f2 a, f2 b)       { f2 s = {-b.y, -b.y}; return a * s; }
#endif

// Packed-over-rows complex rotation: z = x*vd + p*vo for two rows at once.
// cd = (vd_re, vd_im), cv = (vo_re, vo_im). 8 VOP3P ops per two rows.
__device__ __forceinline__ void rotp(f2& zr, f2& zi,
                                     f2 xr, f2 xi, f2 pr, f2 pi, f2 cd, f2 cv) {
  // zr = xr*cd.lo - xi*cd.hi + pr*cv.lo - pi*cv.hi
  f2 t = pk_mul_nbhi(pi, cv);
  t = pk_fma_blo(pr, cv, t);
  t = pk_fma_nbhi(xi, cd, t);
  zr = pk_fma_blo(xr, cd, t);
  // zi = xr*cd.hi + xi*cd.lo + pr*cv.hi + pi*cv.lo
  f2 u = pk_mul_blo(pi, cv);
  u = pk_fma_bhi(pr, cv, u);
  u = pk_fma_blo(xi, cd, u);
  zi = pk_fma_bhi(xr, cd, u);
}

// Thread t owns complex elements 4t..4t+3 of ROWS rows. State is packed over
// row pairs: er[p][k]/ei[p][k] hold (row 2p, row 2p+1) re/im components, so the
// rotation math maps onto v_pk_fma_f32. Even layers: intra-thread pairs.
// Odd layers: boundary exchange via ping-pong LDS (1 barrier each).
// NOTE: placed first in the TU so the disasm snippet shows this kernel.
__global__ __launch_bounds__(TPB) void eunn_apply_kernel(
    const float* __restrict__ x, const float* __restrict__ omega,
    const float4* __restrict__ coeff, float* __restrict__ out, int batch) {
  __shared__ float lds[ROWS * 2 * H];  // 64 KB: async staging, then exchange overlay
  const int t = threadIdx.x;
  const int row0 = blockIdx.x * ROWS;
  const int nrows = min(ROWS, batch - row0);

  f2 er[PAIRS][4] = {}, ei[PAIRS][4] = {};
  const float* src = x + (size_t)row0 * (2 * H);

#if USE_ASYNC
  // Async-stage my 32B per row into LDS. Each lane loads exactly the bytes it
  // reads back, so s_wait_asynccnt alone suffices (no barrier needed yet).
#pragma unroll
  for (int r = 0; r < ROWS; ++r) {
    if (r < nrows) {
      const float* g = src + r * (2 * H) + t * 8;
      float* l = &lds[r * (2 * H) + t * 8];
      __builtin_amdgcn_global_load_async_to_lds_b128((AS_GLOBAL v4i*)g, (AS_LDS v4i*)l, 0, 0);
      __builtin_amdgcn_global_load_async_to_lds_b128((AS_GLOBAL v4i*)(g + 4), (AS_LDS v4i*)(l + 4), 0, 0);
    }
  }
#if __has_builtin(__builtin_amdgcn_s_wait_asynccnt)
  __builtin_amdgcn_s_wait_asynccnt(0);
#else
  asm volatile("s_wait_asynccnt 0" ::: "memory");
#endif
  asm volatile("" ::: "memory");
  // Transposed read: LDS performs the row-major -> packed-over-rows shuffle.
#pragma unroll
  for (int p = 0; p < PAIRS; ++p) {
#pragma unroll
    for (int k = 0; k < 4; ++k) {
      int b0 = (2 * p) * (2 * H) + t * 8 + 2 * k;
      int b1 = (2 * p + 1) * (2 * H) + t * 8 + 2 * k;
      f2 r, i;
      r.x = lds[b0];     r.y = lds[b1];
      i.x = lds[b0 + 1]; i.y = lds[b1 + 1];
      er[p][k] = r; ei[p][k] = i;
    }
  }
#else
#pragma unroll
  for (int r = 0; r < ROWS; ++r) {
    if (r < nrows) {
#pragma unroll
      for (int k = 0; k < 4; ++k) {
        float2 v = ((const float2*)(src + r * (2 * H) + t * 8))[k];
        if (r & 1) { er[r >> 1][k].y = v.x; ei[r >> 1][k].y = v.y; }
        else       { er[r >> 1][k].x = v.x; ei[r >> 1][k].x = v.y; }
      }
    }
  }
#endif
  __syncthreads();  // staging reads done before exchange overlay is written

  // Exchange overlay (64 KB total, same LDS): packed boundary values.
  f2* exRr = (f2*)lds;                   // e3 re, [2][PAIRS][TPB]
  f2* exRi = exRr + 2 * PAIRS * TPB;     // e3 im
  f2* exLr = exRi + 2 * PAIRS * TPB;     // e0 re
  f2* exLi = exLr + 2 * PAIRS * TPB;     // e0 im

  for (int s = 0; s < 16; ++s) {
    // ---- even rotation, layer L = 2s: pairs (e0,e1),(e2,e3) intra-thread ----
    const f2* ce = (const f2*)(coeff + (2 * s) * H + 4 * t);
    f2 d0 = ce[0], v0 = ce[1], d1 = ce[2], v1 = ce[3];
    f2 d2 = ce[4], v2 = ce[5], d3 = ce[6], v3 = ce[7];
#pragma unroll
    for (int p = 0; p < PAIRS; ++p) {
      f2 ar = er[p][0], ai = ei[p][0], br = er[p][1], bi = ei[p][1];
      rotp(er[p][0], ei[p][0], ar, ai, br, bi, d0, v0);
      rotp(er[p][1], ei[p][1], br, bi, ar, ai, d1, v1);
      ar = er[p][2]; ai = ei[p][2]; br = er[p][3]; bi = ei[p][3];
      rotp(er[p][2], ei[p][2], ar, ai, br, bi, d2, v2);
      rotp(er[p][3], ei[p][3], br, bi, ar, ai, d3, v3);
    }
    // ---- odd rotation, layer L = 2s+1: (e1,e2) intra-thread; e0/e3 cross-lane ----
    const f2* co = (const f2*)(coeff + (2 * s + 1) * H + 4 * t);
    d0 = co[0]; v0 = co[1]; d1 = co[2]; v1 = co[3];
    d2 = co[4]; v2 = co[5]; d3 = co[6]; v3 = co[7];
    const int buf = s & 1;  // ping-pong => one barrier per odd layer is enough
#pragma unroll
    for (int p = 0; p < PAIRS; ++p) {
      int slot = (buf * PAIRS + p) * TPB + t;
      exRr[slot] = er[p][3]; exRi[slot] = ei[p][3];
      exLr[slot] = er[p][0]; exLi[slot] = ei[p][0];
    }
    __syncthreads();
#pragma unroll
    for (int p = 0; p < PAIRS; ++p) {
      int slot = (buf * PAIRS + p) * TPB + t;
      // Unconditional neighbor reads: at the row ends (t==0 elem 0, t==255
      // elem 1023) these hit stale/OOB LDS (OOB loads return 0 per ISA), but
      // the odd-layer coefficient there has vo == 0, so the term vanishes.
      f2 lr = exRr[slot - 1];
      f2 li = exRi[slot - 1];
      f2 rr = exLr[slot + 1];
      f2 ri = exLi[slot + 1];
      f2 ar = er[p][1], ai = ei[p][1], br = er[p][2], bi = ei[p][2];
      rotp(er[p][0], ei[p][0], er[p][0], ei[p][0], lr, li, d0, v0);
      rotp(er[p][1], ei[p][1], ar, ai, br, bi, d1, v1);
      rotp(er[p][2], ei[p][2], br, bi, ar, ai, d2, v2);
      rotp(er[p][3], ei[p][3], er[p][3], ei[p][3], rr, ri, d3, v3);
    }
  }

  // ---- final diagonal phase e^{i*omega} (packed over rows) ----
#pragma unroll
  for (int k = 0; k < 4; ++k) {
    float sw, cw;
    sincosf(omega[4 * t + k], &sw, &cw);
    f2 wv = {cw, sw};
#pragma unroll
    for (int p = 0; p < PAIRS; ++p) {
      f2 vr = er[p][k], vi = ei[p][k];
      er[p][k] = pk_fma_blo(vr, wv, pk_mul_nbhi(vi, wv));  // vr*cw - vi*sw
      ei[p][k] = pk_fma_bhi(vr, wv, pk_mul_blo(vi, wv));   // vr*sw + vi*cw
    }
  }

#pragma unroll
  for (int p = 0; p < PAIRS; ++p) {
#pragma unroll
    for (int h = 0; h < 2; ++h) {
      int r = 2 * p + h;
      if (r < nrows) {
        float4* o = (float4*)(out + (size_t)(row0 + r) * (2 * H) + t * 8);
        if (h == 0) {
          o[0] = make_float4(er[p][0].x, ei[p][0].x, er[p][1].x, ei[p][1].x);
          o[1] = make_float4(er[p][2].x, ei[p][2].x, er[p][3].x, ei[p][3].x);
        } else {
          o[0] = make_float4(er[p][0].y, ei[p][0].y, er[p][1].y, ei[p][1].y);
          o[1] = make_float4(er[p][2].y, ei[p][2].y, er[p][3].y, ei[p][3].y);
        }
      }
    }
  }
}

// Precompute per-layer complex coefficients vd/vo for every hidden element.
// Layer L: even rotation if L even (step s = L/2), odd rotation if L odd.
__global__ __launch_bounds__(TPB) void eunn_coeff_kernel(
    const float* __restrict__ even_theta, const float* __restrict__ odd_theta,
    const float* __restrict__ even_phi,  const float* __restrict__ odd_phi,
    float4* __restrict__ coeff) {
  int idx = blockIdx.x * blockDim.x + threadIdx.x;
  if (idx >= LAYERS * H) return;
  int L = idx >> 10;
  int i = idx & (H - 1);
  int s = L >> 1;
  float4 c;
  if ((L & 1) == 0) {                      // even rotation
    int j = i >> 1;
    float st, ct, sp, cp;
    sincosf(even_theta[s * 512 + j], &st, &ct);
    sincosf(even_phi[s * 512 + j], &sp, &cp);
    if ((i & 1) == 0) c = make_float4(sp * ct, cp * ct, -sp * st, -cp * st);
    else              c = make_float4(ct, 0.f, st, 0.f);
  } else {                                 // odd rotation (identity at ends)
    if (i == 0 || i == H - 1) {
      c = make_float4(1.f, 0.f, 0.f, 0.f); // vo = 0 -> partner value irrelevant
    } else {
      int j = (i - 1) >> 1;
      float st, ct, sp, cp;
      sincosf(odd_theta[s * 511 + j], &st, &ct);
      sincosf(odd_phi[s * 511 + j], &sp, &cp);
      if (i & 1) c = make_float4(sp * ct, cp * ct, -sp * st, -cp * st);
      else       c = make_float4(ct, 0.f, st, 0.f);
    }
  }
  coeff[idx] = c;
}

extern "C" void kernel_launch(void* const* d_in, const int* in_sizes, int n_in,
                              void* d_out, int out_size, void* d_ws, size_t ws_size,
                              hipStream_t stream) {
  (void)n_in; (void)out_size; (void)ws_size;
  const float* x          = (const float*)d_in[0];
  const float* omega      = (const float*)d_in[1];
  const float* even_theta = (const float*)d_in[2];
  const float* odd_theta  = (const float*)d_in[3];
  const float* even_phi   = (const float*)d_in[4];
  const float* odd_phi    = (const float*)d_in[5];
  float* out = (float*)d_out;
  float4* coeff = (float4*)d_ws;  // 32*1024*16 B = 512 KB scratch

  int batch = in_sizes[0] / (2 * H);  // 4096

  eunn_coeff_kernel<<<(LAYERS * H + TPB - 1) / TPB, TPB, 0, stream>>>(
      even_theta, odd_theta, even_phi, odd_phi, coeff);

  int grid = (batch + ROWS - 1) / ROWS;
  eunn_apply_kernel<<<grid, TPB, 0, stream>>>(x, omega, coeff, out, batch);
}